// Encoder_57397942944071
// MI455X (gfx1250) — compile-verified
//
#include <hip/hip_runtime.h>
#include <hip/hip_bf16.h>

#define DFEAT 128

typedef __attribute__((ext_vector_type(2))) float v2f;
typedef __attribute__((ext_vector_type(8))) float v8f;

// ---------------------------------------------------------------------------
// Utility: fill a float buffer with a constant (used for deg init + acc zero).
// ---------------------------------------------------------------------------
__global__ void gcn_fill_kernel(float* __restrict__ p, float val, long n) {
    long i = (long)blockIdx.x * blockDim.x + threadIdx.x;
    if (i < n) p[i] = val;
}

// ---------------------------------------------------------------------------
// Degree accumulation: deg[dst[e]] += 1  (deg pre-initialized to 1.0 for the
// implicit self-loop per node).
// ---------------------------------------------------------------------------
__global__ void gcn_deg_kernel(const int* __restrict__ dst, float* __restrict__ deg, int E) {
    int e = blockIdx.x * blockDim.x + threadIdx.x;
    if (e < E) atomicAdd(&deg[dst[e]], 1.0f);
}

// In-place deg -> 1/sqrt(deg). deg >= 1 always (self-loop), matching
// jnp.where(deg > 0, rsqrt(deg), 0).
__global__ void gcn_rsqrt_kernel(float* __restrict__ deg, int N) {
    int i = blockIdx.x * blockDim.x + threadIdx.x;
    if (i < N) deg[i] = rsqrtf(deg[i]);
}

// ---------------------------------------------------------------------------
// H = X @ W, X:[M x 128] fp32, W:[128 x 128] fp32, via V_WMMA_F32_16X16X4_F32.
// One block = 8 wave32; block owns a 16-row M tile; wave w owns N tile w.
// W staged in LDS (64 KB). K=128 -> 32 chained WMMA accumulations.
//
// Fragment layouts per CDNA5 ISA 7.12.2:
//   A (16x4 f32): lanes 0-15 -> M=lane, VGPR{0,1}=K{k,k+1};
//                 lanes 16-31 -> VGPR{0,1}=K{k+2,k+3}
//   B (4x16 f32): lanes 0-15 -> N=lane, VGPR{0,1}=rows K{k,k+1};
//                 lanes 16-31 -> rows K{k+2,k+3}
//   C/D (16x16 f32): VGPR r -> row r (lanes 0-15) / row r+8 (lanes 16-31)
// EXEC must be all ones around WMMA: loads are clamped, never predicated.
// ---------------------------------------------------------------------------
__global__ __launch_bounds__(256) void gcn_gemm128_wmma(const float* __restrict__ X,
                                                        const float* __restrict__ W,
                                                        float* __restrict__ H, int M) {
    __shared__ float Wlds[DFEAT * DFEAT];
    for (int i = threadIdx.x; i < DFEAT * DFEAT; i += 256) Wlds[i] = W[i];
    __syncthreads();

    const int wave = threadIdx.x >> 5;   // 0..7 -> N tile
    const int lane = threadIdx.x & 31;
    const int half = lane >> 4;          // 0: K pair {k,k+1}, 1: {k+2,k+3}
    const int l16  = lane & 15;
    const int n    = wave * 16 + l16;    // output column
    const int row0 = blockIdx.x * 16;

    int arow = row0 + l16;
    if (arow >= M) arow = M - 1;         // clamp: keep EXEC all-ones for WMMA
    const float* __restrict__ Xr = X + (size_t)arow * DFEAT;

    v8f c = {};
#pragma unroll
    for (int k = 0; k < DFEAT; k += 4) {
        const int kb = k + 2 * half;
        v2f a, b;
        a.x = Xr[kb];
        a.y = Xr[kb + 1];
        b.x = Wlds[kb * DFEAT + n];
        b.y = Wlds[(kb + 1) * DFEAT + n];
        c = __builtin_amdgcn_wmma_f32_16x16x4_f32(
                /*neg_a=*/false, a, /*neg_b=*/false, b,
                /*c_mod=*/(short)0, c, /*reuse_a=*/false, /*reuse_b=*/false);
    }

#pragma unroll
    for (int r = 0; r < 8; ++r) {
        const int row = row0 + r + 8 * half;
        if (row < M) H[(size_t)row * DFEAT + n] = c[r];
    }
}

// ---------------------------------------------------------------------------
// Edge scatter: one wave per edge. acc[dst] += H[src] * (dinv[src]*dinv[dst]).
// Lane moves float4 (global_load_b128) + 4x global_atomic_add_f32.
// ---------------------------------------------------------------------------
__global__ __launch_bounds__(256) void gcn_scatter_kernel(const float* __restrict__ H,
                                                          const int* __restrict__ src,
                                                          const int* __restrict__ dst,
                                                          const float* __restrict__ dinv,
                                                          float* __restrict__ acc, int E) {
    const int e = (blockIdx.x << 3) + (threadIdx.x >> 5);
    if (e >= E) return;
    const int lane = threadIdx.x & 31;
    const int s = src[e];
    const int d = dst[e];
    const float nrm = dinv[s] * dinv[d];
    const float4 v = ((const float4*)(H + (size_t)s * DFEAT))[lane];
    float* a = acc + (size_t)d * DFEAT + lane * 4;
    atomicAdd(a + 0, v.x * nrm);
    atomicAdd(a + 1, v.y * nrm);
    atomicAdd(a + 2, v.z * nrm);
    atomicAdd(a + 3, v.w * nrm);
}

// ---------------------------------------------------------------------------
// out = relu(acc + H * dinv^2 (self loop) + bias)   (elementwise, in-place ok)
// ---------------------------------------------------------------------------
__global__ __launch_bounds__(256) void gcn_combine_kernel(const float* __restrict__ acc,
                                                          const float* __restrict__ H,
                                                          const float* __restrict__ dinv,
                                                          const float* __restrict__ bias,
                                                          float* __restrict__ out, long NF) {
    long i = (long)blockIdx.x * blockDim.x + threadIdx.x;
    if (i >= NF) return;
    const int node = (int)(i >> 7);
    const int f    = (int)(i & (DFEAT - 1));
    const float di = dinv[node];
    const float v  = acc[i] + H[i] * di * di + bias[f];
    out[i] = v > 0.0f ? v : 0.0f;
}

// ---------------------------------------------------------------------------
// Orchestration. Inputs: x, edge_index, W1, b1, W2, b2. Output: [N x 128] f32.
// ws layout: dinv (N, padded) | H (N*128) | ACC (N*128). Layer-2 accumulator
// is d_out itself (zeroed here; harness poisons it between runs).
// ---------------------------------------------------------------------------
extern "C" void kernel_launch(void* const* d_in, const int* in_sizes, int n_in,
                              void* d_out, int out_size, void* d_ws, size_t ws_size,
                              hipStream_t stream) {
    const float* X  = (const float*)d_in[0];
    const int*   EI = (const int*)d_in[1];
    const float* W1 = (const float*)d_in[2];
    const float* b1 = (const float*)d_in[3];
    const float* W2 = (const float*)d_in[4];
    const float* b2 = (const float*)d_in[5];
    float* out = (float*)d_out;

    const int N = in_sizes[0] / DFEAT;
    const int E = in_sizes[1] / 2;
    const int* src = EI;
    const int* dst = EI + E;

    float* dinv = (float*)d_ws;                       // N floats (deg -> dinv)
    float* H    = dinv + (((size_t)N + 255) & ~(size_t)255);
    float* ACC  = H + (size_t)N * DFEAT;

    const long NF = (long)N * DFEAT;
    const int nThreads = 256;
    const int nBlkN  = (N + nThreads - 1) / nThreads;
    const int nBlkE  = (E + nThreads - 1) / nThreads;
    const int nBlkNF = (int)((NF + nThreads - 1) / nThreads);
    const int nBlkM  = (N + 15) / 16;
    const int nBlkSc = (E + 7) / 8;   // 8 waves per block, 1 wave per edge

    // Degrees + symmetric normalization
    gcn_fill_kernel<<<nBlkN, nThreads, 0, stream>>>(dinv, 1.0f, (long)N);     // self-loops
    gcn_deg_kernel<<<nBlkE, nThreads, 0, stream>>>(dst, dinv, E);
    gcn_rsqrt_kernel<<<nBlkN, nThreads, 0, stream>>>(dinv, N);

    // Layer 1: H = X @ W1 ; ACC = scatter ; ACC = relu(ACC + self + b1)
    gcn_gemm128_wmma<<<nBlkM, nThreads, 0, stream>>>(X, W1, H, N);
    gcn_fill_kernel<<<nBlkNF, nThreads, 0, stream>>>(ACC, 0.0f, NF);
    gcn_scatter_kernel<<<nBlkSc, nThreads, 0, stream>>>(H, src, dst, dinv, ACC, E);
    gcn_combine_kernel<<<nBlkNF, nThreads, 0, stream>>>(ACC, H, dinv, b1, ACC, NF);

    // Layer 2: H = ACC @ W2 ; out = scatter ; out = relu(out + self + b2)
    gcn_gemm128_wmma<<<nBlkM, nThreads, 0, stream>>>(ACC, W2, H, N);
    gcn_fill_kernel<<<nBlkNF, nThreads, 0, stream>>>(out, 0.0f, NF);
    gcn_scatter_kernel<<<nBlkSc, nThreads, 0, stream>>>(H, src, dst, dinv, out, E);
    gcn_combine_kernel<<<nBlkNF, nThreads, 0, stream>>>(out, H, dinv, b2, out, NF);
}